// HOPE_61392262529218
// MI455X (gfx1250) — compile-verified
//
#include <hip/hip_runtime.h>
#include <math.h>

#define EPSF 1e-5f
#define PIF 3.14159265358979323846f

typedef float v2f __attribute__((ext_vector_type(2)));
typedef float v8f __attribute__((ext_vector_type(8)));

// ---------------------------------------------------------------------------
// Kernel 1: fused front-end. One block per (b,n). 256 threads.
// Computes x = x0 + v * softmax(qk) and writes [B,32,N,K] to xbuf.
// ---------------------------------------------------------------------------
__global__ __launch_bounds__(256) void hope_front(
    const float* __restrict__ dp, const float* __restrict__ new_p,
    const float* __restrict__ exw, const float* __restrict__ exb,
    const float* __restrict__ exg, const float* __restrict__ exbeta,
    const float* __restrict__ exm, const float* __restrict__ exv,
    const float* __restrict__ esw, const float* __restrict__ esb,
    const float* __restrict__ esg, const float* __restrict__ esbeta,
    const float* __restrict__ esm, const float* __restrict__ esv,
    const float* __restrict__ stw, const float* __restrict__ stb,
    const float* __restrict__ stg, const float* __restrict__ stbeta,
    const float* __restrict__ stm, const float* __restrict__ stv,
    const float* __restrict__ k1w, const float* __restrict__ k1b,
    const float* __restrict__ lng, const float* __restrict__ lnb,
    const float* __restrict__ k2w, const float* __restrict__ k2b,
    const float* __restrict__ qw,  const float* __restrict__ qb,
    const float* __restrict__ vw,  const float* __restrict__ vb,
    float* __restrict__ xbuf)
{
    const int bn = blockIdx.x;
    const int b = bn >> 11;        // N = 2048
    const int n = bn & 2047;
    const int tid = threadIdx.x;

    __shared__ float sdp[3][32];
    __shared__ float scat[6][32];
    __shared__ float sx0[32][33];
    __shared__ float ssum[8][3];
    __shared__ int   scnt[8];
    __shared__ float sef[30];
    __shared__ float shr[32];
    __shared__ float skey[32];
    __shared__ float stq[32];
    __shared__ float sa1[32];
    __shared__ float stsc[32], stof[32];

    if (tid < 96) {
        int c = tid >> 5, k = tid & 31;
        sdp[c][k] = dp[((size_t)(b*3 + c)*2048 + n)*32 + k];
    }
    if (tid >= 96 && tid < 128) {
        int d = tid - 96;
        float s = stg[d] * rsqrtf(stv[d] + EPSF);
        stsc[d] = s;
        stof[d] = stbeta[d] - stm[d] * s;
    }
    if (tid >= 128 && tid < 152) { int j = tid - 128; ssum[j/3][j%3] = 0.0f; }
    if (tid >= 152 && tid < 160) scnt[tid - 152] = 0;
    __syncthreads();

    // per-neighbor: embed_xyz + sphere + embed_sphere  (32 threads, one per k)
    if (tid < 32) {
        int k = tid;
        float X = sdp[0][k], Y = sdp[1][k], Z = sdp[2][k];
        #pragma unroll
        for (int c = 0; c < 3; ++c) {
            float y = exw[c*3+0]*X + exw[c*3+1]*Y + exw[c*3+2]*Z + exb[c];
            float s = exg[c] * rsqrtf(exv[c] + EPSF);
            scat[c][k] = (y - exm[c]) * s + exbeta[c];
        }
        float rho = sqrtf(X*X + Y*Y + Z*Z);
        float safe = (rho == 0.0f) ? 1.0f : rho;
        float th = acosf(fminf(fmaxf(Z / safe, -1.0f), 1.0f));
        if (rho == 0.0f) th = 0.0f;
        th *= (1.0f / PIF);
        float ph = atan2f(Y, X) * (1.0f / (2.0f * PIF)) + 0.5f;
        #pragma unroll
        for (int c = 0; c < 3; ++c) {
            float y = esw[c*3+0]*rho + esw[c*3+1]*th + esw[c*3+2]*ph + esb[c];
            float s = esg[c] * rsqrtf(esv[c] + EPSF);
            scat[3+c][k] = (y - esm[c]) * s + esbeta[c];
        }
        // octant accumulation for PointHop
        int oct = ((X > 0.0f) ? 4 : 0) + ((Y > 0.0f) ? 2 : 0) + ((Z > 0.0f) ? 1 : 0);
        atomicAdd(&ssum[oct][0], X);
        atomicAdd(&ssum[oct][1], Y);
        atomicAdd(&ssum[oct][2], Z);
        atomicAdd(&scnt[oct], 1);
    }
    __syncthreads();

    // x0 = relu(bn(stw @ cat6)) : 32x32 values, each thread 4
    for (int idx = tid; idx < 1024; idx += 256) {
        int d = idx >> 5, k = idx & 31;
        float acc = stb[d];
        #pragma unroll
        for (int c = 0; c < 6; ++c) acc += stw[d*6 + c] * scat[c][k];
        sx0[d][k] = fmaxf(acc * stsc[d] + stof[d], 0.0f);
    }

    // ef[30]: std(3) | new_p(3) | octant means(24)
    if (tid >= 32 && tid < 35) {
        int c = tid - 32;
        float mean = 0.0f;
        #pragma unroll
        for (int k = 0; k < 32; ++k) mean += sdp[c][k];
        mean *= (1.0f / 32.0f);
        float ss = 0.0f;
        #pragma unroll
        for (int k = 0; k < 32; ++k) { float d0 = sdp[c][k] - mean; ss += d0*d0; }
        sef[c] = sqrtf(ss * (1.0f / 31.0f));           // ddof=1
        sef[3 + c] = new_p[((size_t)b*2048 + n)*3 + c];
    }
    if (tid >= 64 && tid < 88) {
        int j = tid - 64; int o = j / 3, c = j % 3;
        int cnt = scnt[o];
        sef[6 + j] = (cnt > 0) ? ssum[o][c] / (float)cnt : 0.0f;
    }
    __syncthreads();

    // k_embed: linear(30->32) + LN + relu   (wave 0)
    if (tid < 32) {
        float h = k1b[tid];
        #pragma unroll
        for (int f = 0; f < 30; ++f) h += k1w[tid*30 + f] * sef[f];
        float sum = h;
        #pragma unroll
        for (int off = 16; off > 0; off >>= 1) sum += __shfl_xor(sum, off, 32);
        float mu = sum * (1.0f / 32.0f);
        float dlt = h - mu;
        float sq = dlt * dlt;
        #pragma unroll
        for (int off = 16; off > 0; off >>= 1) sq += __shfl_xor(sq, off, 32);
        float var = sq * (1.0f / 32.0f);               // ddof=0
        float r = rsqrtf(var + EPSF);
        shr[tid] = fmaxf(dlt * r * lng[tid] + lnb[tid], 0.0f);
    }
    __syncthreads();
    if (tid < 32) {
        float acc = k2b[tid];
        #pragma unroll
        for (int d = 0; d < 32; ++d) acc += k2w[tid*32 + d] * shr[d];
        skey[tid] = acc;
    }
    __syncthreads();
    // tq[c] = sum_d key[d]*qw[d][c]
    if (tid < 32) {
        float acc = 0.0f;
        #pragma unroll
        for (int d = 0; d < 32; ++d) acc += skey[d] * qw[d*32 + tid];
        stq[tid] = acc;
    }
    __syncthreads();
    // qkrel + softmax over K (wave 0, lane = k)
    if (tid < 32) {
        float qk = 0.0f;
        #pragma unroll
        for (int c = 0; c < 32; ++c) qk += stq[c] * sx0[c][tid];
        float kb = skey[tid] * qb[tid];
        #pragma unroll
        for (int off = 16; off > 0; off >>= 1) kb += __shfl_xor(kb, off, 32);
        qk += kb;
        float mx = qk;
        #pragma unroll
        for (int off = 16; off > 0; off >>= 1) mx = fmaxf(mx, __shfl_xor(mx, off, 32));
        float e = expf(qk - mx);
        float s = e;
        #pragma unroll
        for (int off = 16; off > 0; off >>= 1) s += __shfl_xor(s, off, 32);
        sa1[tid] = e / s;
    }
    __syncthreads();

    // x = x0 + (vw@x0 + vb) * a1 ; write to xbuf
    for (int idx = tid; idx < 1024; idx += 256) {
        int d = idx >> 5, k = idx & 31;
        float acc = vb[d];
        #pragma unroll
        for (int c = 0; c < 32; ++c) acc += vw[d*32 + c] * sx0[c][k];
        xbuf[((size_t)(b*32 + d)*2048 + n)*32 + k] = sx0[d][k] + acc * sa1[k];
    }
}

// ---------------------------------------------------------------------------
// Kernel 2: eape_pre = relu(bn(m1w[128,32] @ x[32,P])) via V_WMMA_F32_16X16X4
// Block = 256 thr = 8 waves; wave w -> row tile [16w,16w+16); block -> 64 pos.
// ---------------------------------------------------------------------------
__global__ __launch_bounds__(256) void hope_gemm_wmma(
    const float* __restrict__ xbuf, const float* __restrict__ m1w,
    const float* __restrict__ m1b,  const float* __restrict__ m1g,
    const float* __restrict__ m1beta, const float* __restrict__ m1m,
    const float* __restrict__ m1v,  float* __restrict__ eape)
{
    __shared__ float xs[32][80];   // stride 80: rows c and c+2 land 32 banks apart
    __shared__ float fsc[128], fof[128];

    const int tid = threadIdx.x;
    const int blk = blockIdx.x;
    const int b  = blk >> 10;          // 1024 position-tiles per batch
    const int p0 = (blk & 1023) * 64;  // 65536 positions per batch

    if (tid < 128) {
        float s = m1g[tid] * rsqrtf(m1v[tid] + EPSF);
        fsc[tid] = s;
        fof[tid] = m1b[tid] * s + (m1beta[tid] - m1m[tid] * s);
    }
    for (int i = tid; i < 2048; i += 256) {
        int c = i >> 6, pp = i & 63;
        xs[c][pp] = xbuf[(size_t)(b*32 + c)*65536 + p0 + pp];
    }
    __syncthreads();

    const int wv = tid >> 5;
    const int lane = tid & 31;
    const int rowbase = wv * 16;
    const bool hi = lane >= 16;
    const int l16 = lane & 15;

    // A fragments: m1w 16x4 tiles, VGPR0={K0|K2}, VGPR1={K1|K3} per ISA layout
    v2f af[8];
    #pragma unroll
    for (int s = 0; s < 8; ++s) {
        int r  = rowbase + l16;
        int c0 = 4*s + (hi ? 2 : 0);
        af[s][0] = m1w[r*32 + c0];
        af[s][1] = m1w[r*32 + c0 + 1];
    }

    #pragma unroll
    for (int j = 0; j < 4; ++j) {
        const int n0 = j * 16;
        v8f acc = {};
        #pragma unroll
        for (int s = 0; s < 8; ++s) {
            int c0 = 4*s + (hi ? 2 : 0);
            v2f bf;
            bf[0] = xs[c0][n0 + l16];
            bf[1] = xs[c0 + 1][n0 + l16];
            acc = __builtin_amdgcn_wmma_f32_16x16x4_f32(
                false, af[s], false, bf, (short)0, acc, false, false);
        }
        // epilogue: BN affine + ReLU, D layout: VGPR r -> rows r / r+8
        const int n = p0 + n0 + l16;
        #pragma unroll
        for (int r = 0; r < 8; ++r) {
            int o = rowbase + r + (hi ? 8 : 0);
            float v = fmaxf(acc[r] * fsc[o] + fof[o], 0.0f);
            eape[(size_t)(b*128 + o)*65536 + n] = v;
        }
    }
}

// ---------------------------------------------------------------------------
// Kernel 3: gp[b*128+c] = mean over 65536 positions of eape_pre
// ---------------------------------------------------------------------------
__global__ __launch_bounds__(256) void hope_gap(const float* __restrict__ eape,
                                                float* __restrict__ gp)
{
    const int bc = blockIdx.x;
    const int tid = threadIdx.x;
    const float* p = eape + (size_t)bc * 65536;
    float s = 0.0f;
    for (int i = tid; i < 65536; i += 256) s += p[i];
    #pragma unroll
    for (int off = 16; off > 0; off >>= 1) s += __shfl_xor(s, off, 32);
    __shared__ float wsum[8];
    if ((tid & 31) == 0) wsum[tid >> 5] = s;
    __syncthreads();
    if (tid == 0) {
        float t = 0.0f;
        #pragma unroll
        for (int i = 0; i < 8; ++i) t += wsum[i];
        gp[bc] = t * (1.0f / 65536.0f);
    }
}

// ---------------------------------------------------------------------------
// Kernel 4: a2 = sigmoid(conv1d(gp, cmw, k=3, pad=1)) across channel dim
// ---------------------------------------------------------------------------
__global__ __launch_bounds__(256) void hope_gate(const float* __restrict__ gp,
                                                 const float* __restrict__ cmw,
                                                 float* __restrict__ a2)
{
    int i = blockIdx.x * 256 + threadIdx.x;
    if (i >= 1024) return;
    int c = i & 127;
    float l = (c > 0)   ? gp[i - 1] : 0.0f;
    float m = gp[i];
    float r = (c < 127) ? gp[i + 1] : 0.0f;
    float z = cmw[0]*l + cmw[1]*m + cmw[2]*r;
    a2[i] = 1.0f / (1.0f + expf(-z));
}

// ---------------------------------------------------------------------------
// Kernel 5: eape = eape_pre * a2 (in place); f = fj + eape   (float4)
// ---------------------------------------------------------------------------
__global__ __launch_bounds__(256) void hope_final(const float* __restrict__ fj,
                                                  const float* __restrict__ a2,
                                                  float* __restrict__ fout,
                                                  float* __restrict__ eape)
{
    size_t gid = (size_t)blockIdx.x * 256 + threadIdx.x;  // 16,777,216 threads
    size_t idx = gid * 4;
    int bc = (int)(idx >> 16);                            // 65536 floats per (b,c)
    float s = a2[bc];
    float4 e = *(const float4*)(eape + idx);
    float4 f = *(const float4*)(fj + idx);
    e.x *= s; e.y *= s; e.z *= s; e.w *= s;
    f.x += e.x; f.y += e.y; f.z += e.z; f.w += e.w;
    *(float4*)(eape + idx) = e;
    *(float4*)(fout + idx) = f;
}

// ---------------------------------------------------------------------------
extern "C" void kernel_launch(void* const* d_in, const int* in_sizes, int n_in,
                              void* d_out, int out_size, void* d_ws, size_t ws_size,
                              hipStream_t stream)
{
    const float* dp     = (const float*)d_in[0];
    const float* new_p  = (const float*)d_in[1];
    const float* fj     = (const float*)d_in[2];
    // d_in[3] = idx (unused by reference)
    const float* exw    = (const float*)d_in[4];
    const float* exb    = (const float*)d_in[5];
    const float* exg    = (const float*)d_in[6];
    const float* exbeta = (const float*)d_in[7];
    const float* exm    = (const float*)d_in[8];
    const float* exv    = (const float*)d_in[9];
    const float* esw    = (const float*)d_in[10];
    const float* esb    = (const float*)d_in[11];
    const float* esg    = (const float*)d_in[12];
    const float* esbeta = (const float*)d_in[13];
    const float* esm    = (const float*)d_in[14];
    const float* esv    = (const float*)d_in[15];
    const float* stw    = (const float*)d_in[16];
    const float* stb    = (const float*)d_in[17];
    const float* stg    = (const float*)d_in[18];
    const float* stbeta = (const float*)d_in[19];
    const float* stm    = (const float*)d_in[20];
    const float* stv    = (const float*)d_in[21];
    const float* k1w    = (const float*)d_in[22];
    const float* k1b    = (const float*)d_in[23];
    const float* lng    = (const float*)d_in[24];
    const float* lnb    = (const float*)d_in[25];
    const float* k2w    = (const float*)d_in[26];
    const float* k2b    = (const float*)d_in[27];
    const float* qw     = (const float*)d_in[28];
    const float* qb     = (const float*)d_in[29];
    const float* vw     = (const float*)d_in[30];
    const float* vb     = (const float*)d_in[31];
    const float* m1w    = (const float*)d_in[32];
    const float* m1b    = (const float*)d_in[33];
    const float* m1g    = (const float*)d_in[34];
    const float* m1beta = (const float*)d_in[35];
    const float* m1m    = (const float*)d_in[36];
    const float* m1v    = (const float*)d_in[37];
    const float* cmw    = (const float*)d_in[38];

    float* out   = (float*)d_out;
    float* fout  = out;                        // f:    [8,128,2048,32]
    float* eape  = out + (size_t)67108864;     // eape: [8,128,2048,32]
    float* xbuf  = out;                        // scratch x in f region (16.7M floats), overwritten by hope_final
    float* gp    = (float*)d_ws;               // 1024 floats
    float* a2    = gp + 1024;                  // 1024 floats

    hope_front<<<16384, 256, 0, stream>>>(
        dp, new_p, exw, exb, exg, exbeta, exm, exv,
        esw, esb, esg, esbeta, esm, esv,
        stw, stb, stg, stbeta, stm, stv,
        k1w, k1b, lng, lnb, k2w, k2b, qw, qb, vw, vb, xbuf);

    hope_gemm_wmma<<<8192, 256, 0, stream>>>(
        xbuf, m1w, m1b, m1g, m1beta, m1m, m1v, eape);

    hope_gap<<<1024, 256, 0, stream>>>(eape, gp);
    hope_gate<<<4, 256, 0, stream>>>(gp, cmw, a2);
    hope_final<<<65536, 256, 0, stream>>>(fj, a2, fout, eape);
}